// REINFORCE_graph_46643344834924
// MI455X (gfx1250) — compile-verified
//
#include <hip/hip_runtime.h>
#include <math.h>

#define N_NODES 81
#define M_PAD   96
#define N_EDGES 1620
#define IN_DIM  10
#define K_PAD   12
#define HIDDEN  8192
#define ACTIONS 729

typedef __attribute__((ext_vector_type(2))) float v2f;
typedef __attribute__((ext_vector_type(8))) float v8f;

// ---------------- K0: zero accumulators (must run every call) ----------------
__global__ void k0_init(float* g, float* zacc, float* lacc) {
    int i = blockIdx.x * blockDim.x + threadIdx.x;
    if (i < HIDDEN) { g[i] = 0.f; zacc[i] = 0.f; }
    if (i < ACTIONS) lacc[i] = 0.f;
}

// ---------------- K1: build degree-normalized CSR-by-dst edge list ----------
__global__ void k1_graph(const int* __restrict__ ei, float* __restrict__ dinv,
                         int* __restrict__ offs, int* __restrict__ esrc,
                         float* __restrict__ enorm) {
    if (threadIdx.x != 0 || blockIdx.x != 0) return;
    int deg[N_NODES];
    for (int i = 0; i < N_NODES; ++i) deg[i] = 1;            // self loop
    for (int e = 0; e < N_EDGES; ++e) deg[ei[N_EDGES + e]] += 1;
    for (int i = 0; i < N_NODES; ++i) dinv[i] = rsqrtf((float)deg[i]);
    int cur[N_NODES];
    int acc = 0;
    for (int i = 0; i < N_NODES; ++i) { offs[i] = acc; cur[i] = acc; acc += deg[i]; }
    offs[N_NODES] = acc;                                     // = 1701
    for (int e = 0; e < N_EDGES; ++e) {
        int s = ei[e], d = ei[N_EDGES + e];
        int p = cur[d]++;
        esrc[p] = s;
        enorm[p] = dinv[s] * dinv[d];
    }
    for (int i = 0; i < N_NODES; ++i) {                      // self loops
        int p = cur[i]++;
        esrc[p] = i;
        enorm[p] = dinv[i] * dinv[i];
    }
}

// ---------------- K1b: build zero-padded WMMA operands ----------------------
// xp: [96,12] (M and K padded), Wp: [12,8192] (K padded). 98304 threads.
__global__ void k1b_pad(const float* __restrict__ x, const float* __restrict__ W,
                        float* __restrict__ xp, float* __restrict__ Wp) {
    int i = blockIdx.x * blockDim.x + threadIdx.x;           // 0..98303
    int k = i / HIDDEN;                                      // 0..11
    Wp[i] = (k < IN_DIM) ? W[i] : 0.f;
    if (i < M_PAD * K_PAD) {
        int m = i / K_PAD, kk = i % K_PAD;
        xp[i] = (m < N_NODES && kk < IN_DIM) ? x[m * IN_DIM + kk] : 0.f;
    }
}

// ---------------- K2: h = x @ W_gcn via V_WMMA_F32_16X16X4_F32 --------------
// One wave per 16x16 tile; fully padded operands -> branch-free, EXEC all-1s.
__global__ void k2_gcn_mm(const float* __restrict__ xp, const float* __restrict__ Wp,
                          float* __restrict__ h) {
    int lane  = threadIdx.x;          // 0..31
    int ntile = blockIdx.x;           // 0..511
    int mtile = blockIdx.y;           // 0..5
    int khalf = lane >> 4;            // VGPR pair maps K = k0,k0+1 | k0+2,k0+3
    int n = ntile * 16 + (lane & 15);
    int m = mtile * 16 + (lane & 15);
    v8f c = {0.f, 0.f, 0.f, 0.f, 0.f, 0.f, 0.f, 0.f};
#pragma unroll
    for (int kb = 0; kb < K_PAD; kb += 4) {
        int k0 = kb + khalf * 2;
        v2f a, b;
        a.x = xp[m * K_PAD + k0];                // contiguous pair -> b64 load
        a.y = xp[m * K_PAD + k0 + 1];
        b.x = Wp[(size_t)k0 * HIDDEN + n];
        b.y = Wp[(size_t)(k0 + 1) * HIDDEN + n];
        c = __builtin_amdgcn_wmma_f32_16x16x4_f32(false, a, false, b,
                                                  (short)0, c, false, false);
    }
#pragma unroll
    for (int r = 0; r < 8; ++r)                  // padded h: unconditional store
        h[(size_t)(mtile * 16 + r + khalf * 8) * HIDDEN + n] = c[r];
}

// ---------------- K3: symmetric-normalized propagation (gather by dst) ------
__global__ void k3_prop(const float* __restrict__ h, const int* __restrict__ offs,
                        const int* __restrict__ esrc, const float* __restrict__ enorm,
                        float* __restrict__ hp) {
    int col = blockIdx.x * blockDim.x + threadIdx.x;   // 0..8191
    int dst = blockIdx.y;                              // 0..80
    int e0 = offs[dst], e1 = offs[dst + 1];
    float acc = 0.f;
    for (int e = e0; e < e1; ++e)
        acc += enorm[e] * h[(size_t)esrc[e] * HIDDEN + col];
    hp[(size_t)dst * HIDDEN + col] = acc;
}

// ---------------- K4: bias + ReLU + LayerNorm per node, pool into g ---------
__global__ void k4_ln_pool(const float* __restrict__ hp, const float* __restrict__ b_gcn,
                           const float* __restrict__ ln_g, const float* __restrict__ ln_b,
                           float* __restrict__ g) {
    __shared__ float red[256];
    int node = blockIdx.x;
    int tid  = threadIdx.x;
    float t[32];
    float s = 0.f;
#pragma unroll
    for (int i = 0; i < 32; ++i) {
        int j = tid + i * 256;
        float v = hp[(size_t)node * HIDDEN + j] + b_gcn[j];
        v = fmaxf(v, 0.f);
        t[i] = v;
        s += v;
    }
    red[tid] = s; __syncthreads();
    for (int off = 128; off > 0; off >>= 1) {
        if (tid < off) red[tid] += red[tid + off];
        __syncthreads();
    }
    float mu = red[0] * (1.f / (float)HIDDEN);
    __syncthreads();
    float vs = 0.f;
#pragma unroll
    for (int i = 0; i < 32; ++i) { float d = t[i] - mu; vs += d * d; }
    red[tid] = vs; __syncthreads();
    for (int off = 128; off > 0; off >>= 1) {
        if (tid < off) red[tid] += red[tid + off];
        __syncthreads();
    }
    float rinv = rsqrtf(red[0] * (1.f / (float)HIDDEN) + 1e-5f);
#pragma unroll
    for (int i = 0; i < 32; ++i) {
        int j = tid + i * 256;
        float val = (t[i] - mu) * rinv * ln_g[j] + ln_b[j];
        atomicAdd(&g[j], val);
    }
}

// ---------------- K5: zacc = g @ W1 (split-K GEMV, b128 loads) --------------
// grid (8, 128): 1024 cols per block-x, 64 k-rows per block-y
__global__ void k5_gemv1(const float* __restrict__ g, const float* __restrict__ W1,
                         float* __restrict__ zacc) {
    __shared__ float gs[64];
    int tid = threadIdx.x;                 // 256
    int colbase = blockIdx.x * 1024;
    int kbase   = blockIdx.y * 64;
    if (tid < 64) gs[tid] = g[kbase + tid];
    __syncthreads();
    int col = colbase + tid * 4;
    float4 acc = {0.f, 0.f, 0.f, 0.f};
    for (int kk = 0; kk < 64; ++kk) {
        float gv = gs[kk];
        float4 w = *(const float4*)(W1 + (size_t)(kbase + kk) * HIDDEN + col);
        acc.x += gv * w.x; acc.y += gv * w.y;
        acc.z += gv * w.z; acc.w += gv * w.w;
    }
    atomicAdd(&zacc[col + 0], acc.x);
    atomicAdd(&zacc[col + 1], acc.y);
    atomicAdd(&zacc[col + 2], acc.z);
    atomicAdd(&zacc[col + 3], acc.w);
}

// ---------------- K6: lacc = relu(z + b1) @ W2 (split-K GEMV) ---------------
// grid 64: 128 k-rows per block, all 729 cols (3 cols/thread)
__global__ void k6_gemv2(const float* __restrict__ zacc, const float* __restrict__ b1,
                         const float* __restrict__ W2, float* __restrict__ lacc) {
    __shared__ float zs[128];
    int tid = threadIdx.x;                 // 256
    int kbase = blockIdx.x * 128;
    if (tid < 128) zs[tid] = fmaxf(zacc[kbase + tid] + b1[kbase + tid], 0.f);
    __syncthreads();
    int c0 = tid, c1 = tid + 256, c2 = tid + 512;
    float a0 = 0.f, a1 = 0.f, a2 = 0.f;
    for (int kk = 0; kk < 128; ++kk) {
        float z = zs[kk];
        const float* row = W2 + (size_t)(kbase + kk) * ACTIONS;
        a0 += z * row[c0];
        a1 += z * row[c1];
        if (c2 < ACTIONS) a2 += z * row[c2];
    }
    atomicAdd(&lacc[c0], a0);
    atomicAdd(&lacc[c1], a1);
    if (c2 < ACTIONS) atomicAdd(&lacc[c2], a2);
}

// ---------------- K7: log_softmax over 729 logits ---------------------------
__global__ void k7_logsoftmax(const float* __restrict__ lacc, const float* __restrict__ b2,
                              float* __restrict__ out) {
    __shared__ float red[1024];
    int tid = threadIdx.x;                 // 1024
    float v = (tid < ACTIONS) ? lacc[tid] + b2[tid] : -3.402823466e38f;
    red[tid] = v; __syncthreads();
    for (int off = 512; off > 0; off >>= 1) {
        if (tid < off) red[tid] = fmaxf(red[tid], red[tid + off]);
        __syncthreads();
    }
    float mx = red[0];
    __syncthreads();
    float e = (tid < ACTIONS) ? expf(v - mx) : 0.f;
    red[tid] = e; __syncthreads();
    for (int off = 512; off > 0; off >>= 1) {
        if (tid < off) red[tid] += red[tid + off];
        __syncthreads();
    }
    float lse = logf(red[0]);
    if (tid < ACTIONS) out[tid] = (v - mx) - lse;
}

extern "C" void kernel_launch(void* const* d_in, const int* in_sizes, int n_in,
                              void* d_out, int out_size, void* d_ws, size_t ws_size,
                              hipStream_t stream) {
    const float* x     = (const float*)d_in[0];
    const int*   ei    = (const int*)  d_in[1];
    const float* W_gcn = (const float*)d_in[2];
    const float* b_gcn = (const float*)d_in[3];
    const float* ln_g  = (const float*)d_in[4];
    const float* ln_b  = (const float*)d_in[5];
    const float* W1    = (const float*)d_in[6];
    const float* b1    = (const float*)d_in[7];
    const float* W2    = (const float*)d_in[8];
    const float* b2    = (const float*)d_in[9];
    float* out = (float*)d_out;

    char* ws = (char*)d_ws;
    float* dinv  = (float*)(ws + 0);        // 81 f
    int*   offs  = (int*)  (ws + 512);      // 82 i
    int*   esrc  = (int*)  (ws + 1024);     // 1701 i
    float* enorm = (float*)(ws + 8192);     // 1701 f
    float* xp    = (float*)(ws + 16384);    // 96*12 f
    float* Wp    = (float*)(ws + 24576);    // 12*8192 f
    float* h     = Wp + (size_t)K_PAD * HIDDEN;     // 96*8192 f (padded M)
    float* hp    = h  + (size_t)M_PAD * HIDDEN;     // 81*8192 f
    float* g     = hp + (size_t)N_NODES * HIDDEN;   // 8192 f
    float* zacc  = g  + HIDDEN;                     // 8192 f
    float* lacc  = zacc + HIDDEN;                   // 729 f

    k0_init      <<<32, 256, 0, stream>>>(g, zacc, lacc);
    k1_graph     <<<1, 1, 0, stream>>>(ei, dinv, offs, esrc, enorm);
    k1b_pad      <<<(K_PAD * HIDDEN) / 256, 256, 0, stream>>>(x, W_gcn, xp, Wp);
    k2_gcn_mm    <<<dim3(HIDDEN / 16, 6), 32, 0, stream>>>(xp, Wp, h);
    k3_prop      <<<dim3(HIDDEN / 256, N_NODES), 256, 0, stream>>>(h, offs, esrc, enorm, hp);
    k4_ln_pool   <<<N_NODES, 256, 0, stream>>>(hp, b_gcn, ln_g, ln_b, g);
    k5_gemv1     <<<dim3(8, 128), 256, 0, stream>>>(g, W1, zacc);
    k6_gemv2     <<<64, 256, 0, stream>>>(zacc, b1, W2, lacc);
    k7_logsoftmax<<<1, 1024, 0, stream>>>(lacc, b2, out);
}